// RadiusInteractionGraph_63273458204901
// MI455X (gfx1250) — compile-verified
//
#include <hip/hip_runtime.h>
#include <math.h>

// radius_graph(pos, r=10, batch sorted 16x1024, K=32) on gfx1250.
//
// d^2 tile (16x16) computed in ONE V_WMMA_F32_16X16X4_F32:
//   A[m] = (-2x_m, -2y_m, -2z_m, 1),  B[:,n] = (x_n, y_n, z_n, |p_n|^2),
//   C[m][n] = |p_m|^2   =>   D[m][n] = d^2(m,n).
// One wave per (graph, 16-row tile): 64 WMMAs -> 16x1024 d^2 strip in LDS,
// then lanes 0..15 do a register-resident sorted top-32 scan per row.
// Self-loop masking applied only on the diagonal tile (ct == rt).

typedef float v2f __attribute__((ext_vector_type(2)));
typedef float v8f __attribute__((ext_vector_type(8)));

#define NPTS 16384
#define KNN 32
#define R2CUT 100.0f      // cutoff^2
#define BIGF 1e30f

__global__ __launch_bounds__(32)
void RadiusGraph_wmma_kernel(const float* __restrict__ pos,
                             float* __restrict__ out_col,
                             float* __restrict__ out_row,
                             float* __restrict__ out_w,
                             float* __restrict__ out_m)
{
    __shared__ float strip[16 * 1024];   // d^2 strip: 16 rows x 1024 cols (64 KB)

    const int lane = threadIdx.x;        // wave32
    const int half = lane >> 4;          // 0: lanes 0-15, 1: lanes 16-31
    const int l16  = lane & 15;

    const int g  = blockIdx.x >> 6;      // graph id (16 graphs)
    const int rt = blockIdx.x & 63;      // 16-row tile within graph (64 tiles)
    const int r0 = (g << 10) + (rt << 4);// global first row of this tile

    // ---- A fragment (constant across all column tiles) ----
    // lanes 0-15:  VGPR0 = A[M=l16][K=0] = -2x, VGPR1 = A[M][K=1] = -2y
    // lanes 16-31: VGPR0 = A[M=l16][K=2] = -2z, VGPR1 = A[M][K=3] =  1
    const int arow = r0 + l16;
    const float ax = pos[3 * arow + 0];
    const float ay = pos[3 * arow + 1];
    const float az = pos[3 * arow + 2];
    v2f a;
    a.x = half ? (-2.0f * az) : (-2.0f * ax);
    a.y = half ? 1.0f         : (-2.0f * ay);

    // ---- C init: C[m][n] = |p_m|^2 ; vgpr v holds row (v + 8*half) ----
    v8f cinit;
#pragma unroll
    for (int v = 0; v < 8; ++v) {
        const int rr = r0 + v + (half << 3);
        const float x = pos[3 * rr + 0];
        const float y = pos[3 * rr + 1];
        const float z = pos[3 * rr + 2];
        cinit[v] = x * x + y * y + z * z;
    }

    // ---- Phase 1: 64 column tiles, one WMMA each ----
    for (int ct = 0; ct < 64; ++ct) {
        const int j0 = (g << 10) + (ct << 4);       // global first col of tile
        const int bn = j0 + l16;
        const float bx = pos[3 * bn + 0];
        const float by = pos[3 * bn + 1];
        const float bz = pos[3 * bn + 2];
        const float bp2 = bx * bx + by * by + bz * bz;
        // lanes 0-15:  VGPR0 = B[K=0][N=l16] = x, VGPR1 = B[K=1][N] = y
        // lanes 16-31: VGPR0 = B[K=2][N=l16] = z, VGPR1 = B[K=3][N] = |p|^2
        v2f b;
        b.x = half ? bz  : bx;
        b.y = half ? bp2 : by;

        v8f c = cinit;
        // 8 args: (neg_a, A, neg_b, B, c_mod, C, reuse_a, reuse_b)
        c = __builtin_amdgcn_wmma_f32_16x16x4_f32(
                false, a, false, b, (short)0, c, false, false);

        const int jloc = (ct << 4) + l16;           // local col 0..1023
        const bool diag = (ct == rt);               // wave-uniform

        if (diag) {
            // self-pairs live only here: row (v + 8*half) == col l16
#pragma unroll
            for (int v = 0; v < 8; ++v) {
                const int rloc = v + (half << 3);
                float d2 = fmaxf(c[v], 0.0f);
                if (l16 == rloc) d2 = BIGF;         // kill self-loop
                strip[(rloc << 10) + jloc] = d2;
            }
        } else {
#pragma unroll
            for (int v = 0; v < 8; ++v) {
                const int rloc = v + (half << 3);
                strip[(rloc << 10) + jloc] = fmaxf(c[v], 0.0f);
            }
        }
    }
    __syncthreads();

    // ---- Phase 2: lanes 0..15 each do top-32 over one row ----
    if (lane < 16) {
        const int m = lane;
        const int i = r0 + m;                        // global node id

        float bd[KNN];
        int   bj[KNN];
#pragma unroll
        for (int k = 0; k < KNN; ++k) { bd[k] = BIGF; bj[k] = i; }

        const float4* rowp = (const float4*)(&strip[m << 10]);
        for (int q = 0; q < 256; ++q) {
            const int qq = (q + m) & 255;            // stagger: spread LDS banks
            const float4 dv = rowp[qq];
            const int jb = (g << 10) + (qq << 2);    // global col of dv.x
            const float dd[4] = { dv.x, dv.y, dv.z, dv.w };
#pragma unroll
            for (int t = 0; t < 4; ++t) {
                float cd = dd[t];
                if (cd <= R2CUT && cd < bd[KNN - 1]) {
                    int cj = jb + t;
#pragma unroll
                    for (int k = 0; k < KNN; ++k) {  // branchless sorted insert
                        const bool sm = cd < bd[k];
                        const float nd = sm ? bd[k] : cd;
                        const int   nj = sm ? bj[k] : cj;
                        bd[k] = sm ? cd : bd[k];
                        bj[k] = sm ? cj : bj[k];
                        cd = nd; cj = nj;
                    }
                }
            }
        }

        // ---- emit edges: ascending distance, PyG (col=source, row=target) ----
        const int base = i * KNN;
#pragma unroll
        for (int k = 0; k < KNN; ++k) {
            const float dk = bd[k];
            const bool valid = dk <= R2CUT;
            out_col[base + k] = (float)bj[k];
            out_row[base + k] = (float)i;
            out_w[base + k]   = valid ? sqrtf(dk) : 0.0f;
            out_m[base + k]   = valid ? 1.0f : 0.0f;
        }
    }
}

extern "C" void kernel_launch(void* const* d_in, const int* in_sizes, int n_in,
                              void* d_out, int out_size, void* d_ws, size_t ws_size,
                              hipStream_t stream) {
    (void)in_sizes; (void)n_in; (void)d_ws; (void)ws_size; (void)out_size;
    const float* pos = (const float*)d_in[0];
    // d_in[1] (batch) is structurally 16 contiguous graphs of 1024 nodes.
    float* out = (float*)d_out;
    const int NK = NPTS * KNN;
    RadiusGraph_wmma_kernel<<<dim3(NPTS / 16), dim3(32), 0, stream>>>(
        pos, out, out + NK, out + 2 * NK, out + 3 * NK);
}